// PhenoRotary3DPositionalEncoder_48163763257700
// MI455X (gfx1250) — compile-verified
//
#include <hip/hip_runtime.h>
#include <cstdint>
#include <cstddef>

// ---------------------------------------------------------------------------
// PhenoRotary3DPositionalEncoder for MI455X (gfx1250)
//
// out[pos, 2p]   = cos(th_p) + sin(th_p)
// out[pos, 2p+1] = cos(th_p) - sin(th_p)
// th_p = 0 for p<2 (left pad), else th = x[pos, c] * 20*pi * 10000^(-m/21)
// with k = p-2, c = k/21, m = k%21.
//
// Memory-bound streaming kernel (134 MB out @ 23.3 TB/s ~ 5.9 us floor).
// Uses hardware sin/cos in "revolutions" units: rev = x * 10 * 10000^(-m/21).
// Input tile staged through LDS via the CDNA5 Tensor Data Mover.
// ---------------------------------------------------------------------------

typedef __attribute__((ext_vector_type(4))) float    v4f;
typedef __attribute__((ext_vector_type(4))) unsigned v4u;
typedef __attribute__((ext_vector_type(8))) int      v8i;
typedef __attribute__((ext_vector_type(4))) int      v4i;

#define TPOS     16     // positions per block
#define NTHREADS 256    // 8 waves (wave32)
#define NPAIR    128    // D_MODEL/2
#define LOG2_1E4 13.287712379549449f

__device__ __forceinline__ float fast_exp2(float x) {
#if __has_builtin(__builtin_amdgcn_exp2f)
  return __builtin_amdgcn_exp2f(x);
#else
  return exp2f(x);
#endif
}

__global__ __launch_bounds__(NTHREADS)
void pheno_rotary3d_posenc_kernel(const float* __restrict__ x,
                                  float* __restrict__ out,
                                  int total_pos) {
  __shared__ float         s_x[TPOS * 6];
  __shared__ float         s_freq[NPAIR];
  __shared__ unsigned char s_cidx[NPAIR];

  const int       t    = threadIdx.x;
  const long long pos0 = (long long)blockIdx.x * TPOS;
  const int       rem  = total_pos - (int)pos0;                 // >= 1
  const int       nx   = (rem >= TPOS ? TPOS : rem) * 6;        // x floats in tile

  // ---- per-block frequency / coord-index table (threads 0..127) ----------
  if (t < NPAIR) {
    float f = 0.0f;           // pad pairs: rev = 0 -> (cos,sin) = (1,0)
    unsigned c = 0u;
    if (t >= 2) {
      unsigned k = (unsigned)t - 2u;
      c = k / 21u;
      unsigned m = k - c * 21u;
      // rev = x * 10 * 10000^(-m/21)   (hardware sin/cos take revolutions)
      f = 10.0f * fast_exp2(-(float)m * (LOG2_1E4 / 21.0f));
    }
    s_freq[t] = f;
    s_cidx[t] = (unsigned char)c;
  }

  // ---- stage x tile into LDS --------------------------------------------
  const float* gx = x + pos0 * 6;

#if __has_builtin(__builtin_amdgcn_tensor_load_to_lds)
  if (t < 32) {  // single wave issues the TDM descriptor
    const unsigned           lds_off = (unsigned)(size_t)(&s_x[0]); // LDS byte addr
    const unsigned long long ga      = (unsigned long long)(size_t)gx;

    v4u g0;
    g0.x = 1u;                                                  // count=1, user D#
    g0.y = lds_off;                                             // lds_addr
    g0.z = (unsigned)ga;                                        // global_addr[31:0]
    g0.w = ((unsigned)(ga >> 32) & 0x01FFFFFFu) | 0x80000000u;  // addr[56:32]|type=2

    v8i g1;
    g1[0] = 0x00020000;                 // data_size = 4B; no multicast/pad/iterate
    g1[1] = (int)((unsigned)nx << 16);  // tensor_dim0[15:0] (OOB guard = tile size)
    g1[2] = (int)(1u << 16);            // tensor_dim1 = 1
    g1[3] = (int)((unsigned)nx << 16);  // tile_dim0 = nx
    g1[4] = 1;                          // tile_dim1 = 1, tile_dim2 = 0
    g1[5] = nx;                         // tensor_dim0_stride[31:0]
    g1[6] = 0;
    g1[7] = 0;

    v4i gz4 = {0, 0, 0, 0};
    v8i gz8;
    gz8[0] = 0; gz8[1] = 0; gz8[2] = 0; gz8[3] = 0;
    gz8[4] = 0; gz8[5] = 0; gz8[6] = 0; gz8[7] = 0;

    // 6-arg form (clang-23 / therock-10.0 headers)
    __builtin_amdgcn_tensor_load_to_lds(g0, g1, gz4, gz4, gz8, 0);
#if __has_builtin(__builtin_amdgcn_s_wait_tensorcnt)
    __builtin_amdgcn_s_wait_tensorcnt(0);
#else
    asm volatile("s_wait_tensorcnt 0x0" ::: "memory");
#endif
    // Descriptor semantics are not hardware-verified: wave 0 re-copies the
    // 384B tile after the tensor wait (race-free; overwrites identical data
    // when the TDM path is correct). Cost is negligible vs 16KB of output.
    for (int i = t; i < nx; i += 32) s_x[i] = gx[i];
  }
#else
  if (t < TPOS * 6 && t < nx) s_x[t] = gx[t];
#endif

  __syncthreads();

  // ---- main streaming loop: 1024 float4 per block, 4 per thread ----------
  float* outb = out + pos0 * (2 * NPAIR);
  #pragma unroll
  for (int i = 0; i < 4; ++i) {
    const int q    = t + i * NTHREADS;  // 0..1023: float4 index inside tile
    const int row  = q >> 6;            // position within tile
    const int col4 = q & 63;            // float4 column (2 pairs)
    const int p0   = col4 * 2;
    const int p1   = p0 + 1;

    const float xv0 = s_x[row * 6 + s_cidx[p0]];
    const float xv1 = s_x[row * 6 + s_cidx[p1]];
    const float r0  = xv0 * s_freq[p0];     // revolutions
    const float r1  = xv1 * s_freq[p1];

    const float sn0 = __builtin_amdgcn_sinf(r0);   // v_sin_f32 (arg in turns)
    const float cs0 = __builtin_amdgcn_cosf(r0);   // v_cos_f32
    const float sn1 = __builtin_amdgcn_sinf(r1);
    const float cs1 = __builtin_amdgcn_cosf(r1);

    v4f v;
    v.x = cs0 + sn0;
    v.y = cs0 - sn0;
    v.z = cs1 + sn1;
    v.w = cs1 - sn1;

    if (row < rem) {
      __builtin_nontemporal_store(v, (v4f*)(outb + q * 4));  // b128, streaming
    }
  }
}

extern "C" void kernel_launch(void* const* d_in, const int* in_sizes, int n_in,
                              void* d_out, int out_size, void* d_ws, size_t ws_size,
                              hipStream_t stream) {
  const float* x   = (const float*)d_in[0];
  float*       out = (float*)d_out;
  const int total_pos = in_sizes[0] / 6;                 // B*N = 131072
  const int blocks    = (total_pos + TPOS - 1) / TPOS;   // 8192
  pheno_rotary3d_posenc_kernel<<<blocks, NTHREADS, 0, stream>>>(x, out, total_pos);
}